// DGCNN_45810121179449
// MI455X (gfx1250) — compile-verified
//
#include <hip/hip_runtime.h>

#define EPSV 1e-5f
#define KNN 20

// Set to 0 if the gfx1250 assembler rejects the async-to-LDS mnemonics.
#define USE_ASYNC_COPY 1

typedef float v2f __attribute__((ext_vector_type(2)));
typedef float v8f __attribute__((ext_vector_type(8)));

// D = A(16x4,f32) * B(4x16,f32) + C(16x16,f32)  -- CDNA5 V_WMMA_F32_16X16X4_F32
__device__ __forceinline__ v8f wmma4(v2f a, v2f b, v8f c) {
    return __builtin_amdgcn_wmma_f32_16x16x4_f32(
        /*neg_a=*/false, a, /*neg_b=*/false, b,
        /*c_mod=*/(short)0, c, /*reuse_a=*/false, /*reuse_b=*/false);
}

// 16B global -> LDS direct copy (ASYNCcnt-tracked; no VGPR round trip).
__device__ __forceinline__ void async_cp16(float* lds_dst, const float* gsrc) {
#if USE_ASYNC_COPY
    unsigned lo = (unsigned)(uintptr_t)lds_dst;   // LDS byte offset (low 32 bits)
    asm volatile("global_load_async_to_lds_b128 %0, %1, off"
                 :: "v"(lo), "v"(gsrc) : "memory");
#else
    float4 v = *(const float4*)gsrc;
    lds_dst[0] = v.x; lds_dst[1] = v.y; lds_dst[2] = v.z; lds_dst[3] = v.w;
#endif
}

__device__ __forceinline__ void async_cp_fence() {
#if USE_ASYNC_COPY
    asm volatile("s_wait_asynccnt 0x0" ::: "memory");
#endif
}

// ---------------------------------------------------------------------------
// 1) per-row squared norms:  xx[r] = sum_c X[r][c]^2
// ---------------------------------------------------------------------------
__global__ void rowsq_kernel(const float* __restrict__ X, int ld, int C,
                             float* __restrict__ xx, int rows) {
    int r = blockIdx.x * blockDim.x + threadIdx.x;
    if (r >= rows) return;
    const float* p = X + (size_t)r * ld;
    float s = 0.f;
    if ((C & 3) == 0) {
        for (int c = 0; c < C; c += 4) {
            float4 v = *(const float4*)(p + c);
            s += v.x * v.x + v.y * v.y + v.z * v.z + v.w * v.w;
        }
    } else {
        for (int c = 0; c < C; ++c) { float t = p[c]; s += t * t; }
    }
    xx[r] = s;
}

// ---------------------------------------------------------------------------
// 2) fused knn: 4 waves/block, each wave owns 16 query rows; the 16-column
//    tile is staged ONCE per block (async global->LDS for aligned C).
//    Inner products via f32 WMMA. Both half-waves keep a top-20 list per row
//    (8 columns each per tile); the two sorted lists are merged at the end.
// ---------------------------------------------------------------------------
template <int C>
__global__ __launch_bounds__(128)
void knn_kernel(const float* __restrict__ X, int ld,
                const float* __restrict__ xx,
                int* __restrict__ idx, int N) {
    constexpr int KS  = (C + 3) / 4;
    constexpr int LDB = (C % 4 == 0) ? (C + 4) : (C + 3); // 16B-aligned rows
    __shared__ float ldsB[16 * LDB];
    __shared__ float ldsXX[16];
    __shared__ float ldsD[4][16 * 20];                    // column-major, b128 stores
    __shared__ float mrgV[4][32][KNN];
    __shared__ int   mrgI[4][32][KNN];

    const int tid  = threadIdx.x;
    const int wave = tid >> 5;
    const int lane = tid & 31;
    const int r16  = lane & 15;
    const int hi   = lane >> 4;
    const int b    = blockIdx.y;
    const int rowbase = blockIdx.x * 64 + wave * 16;
    const int grow = b * N + rowbase + r16;

    // Preload A fragments (this wave's 16 query rows).
    v2f areg[KS];
    const float* __restrict__ Arow = X + (size_t)grow * ld;
#pragma unroll
    for (int s = 0; s < KS; ++s) {
        int k0 = s * 4 + hi * 2;
        if constexpr (C % 4 == 0) {
            areg[s] = *(const v2f*)(Arow + k0);            // 8B-aligned (ld even)
        } else {
            areg[s].x = (k0     < C) ? Arow[k0]     : 0.f;
            areg[s].y = (k0 + 1 < C) ? Arow[k0 + 1] : 0.f;
        }
    }
    const float xxr = xx[grow];

    float topv[KNN]; int topi[KNN];
#pragma unroll
    for (int q = 0; q < KNN; ++q) { topv[q] = -3.0e38f; topi[q] = 0; }

    for (int jt = 0; jt < N / 16; ++jt) {
        const int colbase = jt * 16;
        const float* __restrict__ Xc = X + (size_t)(b * N + colbase) * ld;
        if constexpr (C % 4 == 0) {
            // async copy: 16 rows x C floats, 16B per lane-op, no tail divergence
            for (int t = tid; t < 16 * (C / 4); t += 128) {
                int cc = t / (C / 4), k4 = (t % (C / 4)) * 4;
                async_cp16(&ldsB[cc * LDB + k4], Xc + (size_t)cc * ld + k4);
            }
        } else {
            for (int t = tid; t < 16 * C; t += 128) {
                int cc = t / C, kk = t % C;
                ldsB[cc * LDB + kk] = Xc[(size_t)cc * ld + kk];
            }
            // zero-fill padding so unguarded b64 fragment reads are safe
            constexpr int PAD = LDB - C;
            for (int t = tid; t < 16 * PAD; t += 128) {
                int cc = t / PAD, kk = C + (t % PAD);
                ldsB[cc * LDB + kk] = 0.f;
            }
        }
        if (tid < 16) ldsXX[tid] = xx[b * N + colbase + tid];
        async_cp_fence();          // own wave's async writes visible
        __syncthreads();           // ...then all waves'

        v8f acc = {};
#pragma unroll
        for (int s = 0; s < KS; ++s) {
            int k0 = s * 4 + hi * 2;
            v2f bf = *(const v2f*)&ldsB[r16 * LDB + k0];   // 8B-aligned
            acc = wmma4(areg[s], bf, acc);
        }
        // D: reg v, lane L -> (row = v + 8*hi, col = r16). One lane owns 8
        // contiguous rows of one column -> two 16B LDS stores (column-major).
        {
            float4* dst = (float4*)&ldsD[wave][r16 * 20 + 8 * hi];
            dst[0] = ((const float4*)&acc)[0];
            dst[1] = ((const float4*)&acc)[1];
        }

        // scan: lane handles row r16; half-waves split the 16 columns.
        // Keep rolled: the insertion chain appears once in the hot loop.
#pragma unroll 1
        for (int c = 0; c < 8; ++c) {
            int cc = hi * 8 + c;
            float val = 2.f * ldsD[wave][cc * 20 + r16] - xxr - ldsXX[cc];
            if (val > topv[KNN - 1]) {
                topv[KNN - 1] = val; topi[KNN - 1] = colbase + cc;
#pragma unroll
                for (int q = KNN - 1; q > 0; --q) {
                    if (topv[q] > topv[q - 1]) {
                        float tv = topv[q]; topv[q] = topv[q-1]; topv[q-1] = tv;
                        int   ti = topi[q]; topi[q] = topi[q-1]; topi[q-1] = ti;
                    }
                }
            }
        }
        __syncthreads();   // protect ldsB/ldsXX before next staging
    }

    // merge the two sorted top-20 lists per row (lanes r and r+16)
#pragma unroll
    for (int q = 0; q < KNN; ++q) { mrgV[wave][lane][q] = topv[q]; mrgI[wave][lane][q] = topi[q]; }
    __syncthreads();
    if (lane < 16) {
        int ia = 0, ib = 0;
        int* op = idx + (size_t)grow * KNN;
#pragma unroll
        for (int q = 0; q < KNN; ++q) {
            float va = mrgV[wave][r16][ia];
            float vb = mrgV[wave][16 + r16][ib];
            bool ta = (va >= vb);
            op[q] = ta ? mrgI[wave][r16][ia] : mrgI[wave][16 + r16][ib];
            ia += ta ? 1 : 0;
            ib += ta ? 0 : 1;
        }
    }
}

// ---------------------------------------------------------------------------
// 3) fused edge-conv: one wave per point. Edge features (32 x 2C, rows>=20
//    and padded K columns zeroed) staged in LDS; per 16-col output tile:
//    2 row-tile WMMAs, fused bias + BN + ReLU, max over the 20 valid rows.
// ---------------------------------------------------------------------------
template <int C, int O>
__global__ __launch_bounds__(32)
void edgeconv_kernel(const float* __restrict__ X, int ld,
                     const int* __restrict__ idx,
                     const float* __restrict__ W,    // (O, 2C)
                     const float* __restrict__ cb,
                     const float* __restrict__ g,
                     const float* __restrict__ be,
                     const float* __restrict__ m,
                     const float* __restrict__ vv,
                     float* __restrict__ out, int ldo, int N) {
    constexpr int KD  = 2 * C;
    constexpr int KS  = (KD + 3) / 4;
    constexpr int LDE = KD + 2;                     // even stride, >= 4*KS
    __shared__ float E[32 * LDE];

    const int lane = threadIdx.x;
    const int gpt  = blockIdx.x;                    // point id in [0, B*N)
    const int b    = gpt / N;
    const int r16  = lane & 15;
    const int hi   = lane >> 4;

    const float* __restrict__ xi = X + (size_t)gpt * ld;
    const int*   __restrict__ nb = idx + (size_t)gpt * KNN;

    if constexpr (C % 4 == 0) {
        constexpr int K4 = KD / 4;
        for (int t = lane; t < 32 * K4; t += 32) {
            int rr = t / K4, k4 = (t % K4) * 4;
            float4 v = make_float4(0.f, 0.f, 0.f, 0.f);
            if (rr < KNN) {
                if (k4 < C) {                       // neighbor - center
                    int j = nb[rr];
                    float4 a = *(const float4*)(X + (size_t)(b * N + j) * ld + k4);
                    float4 c = *(const float4*)(xi + k4);
                    v = make_float4(a.x - c.x, a.y - c.y, a.z - c.z, a.w - c.w);
                } else {                            // center
                    v = *(const float4*)(xi + k4 - C);
                }
            }
            E[rr * LDE + k4 + 0] = v.x;
            E[rr * LDE + k4 + 1] = v.y;
            E[rr * LDE + k4 + 2] = v.z;
            E[rr * LDE + k4 + 3] = v.w;
        }
        for (int t = lane; t < 32 * (LDE - KD); t += 32) {
            int rr = t / (LDE - KD), kk = KD + (t % (LDE - KD));
            E[rr * LDE + kk] = 0.f;
        }
    } else {
        for (int t = lane; t < 32 * LDE; t += 32) {
            int rr = t / LDE, kk = t % LDE;
            float val = 0.f;
            if (rr < KNN && kk < KD) {
                if (kk < C) {
                    int j = nb[rr];
                    val = X[(size_t)(b * N + j) * ld + kk] - xi[kk];
                } else {
                    val = xi[kk - C];
                }
            }
            E[rr * LDE + kk] = val;
        }
    }
    __syncthreads();

    for (int ot = 0; ot < O / 16; ++ot) {
        const int o = ot * 16 + r16;
        const float* __restrict__ Wr = W + (size_t)o * KD;
        __builtin_prefetch(Wr, 0, 1);               // global_prefetch_b8
        v8f acc0 = {}, acc1 = {};
#pragma unroll
        for (int s = 0; s < KS; ++s) {
            int k0 = s * 4 + hi * 2;
            v2f a0 = *(const v2f*)&E[r16 * LDE + k0];        // 8B-aligned
            v2f a1 = *(const v2f*)&E[(16 + r16) * LDE + k0];
            v2f bf;
            if constexpr (KD % 4 == 0) {
                bf = *(const v2f*)(Wr + k0);
            } else {
                bf.x = (k0     < KD) ? Wr[k0]     : 0.f;
                bf.y = (k0 + 1 < KD) ? Wr[k0 + 1] : 0.f;
            }
            acc0 = wmma4(a0, bf, acc0);                      // rows 0..15
            acc1 = wmma4(a1, bf, acc1);                      // rows 16..31
        }
        const float sc = g[o] * rsqrtf(vv[o] + EPSV);
        const float sh = be[o] + (cb[o] - m[o]) * sc;
        float mval = 0.f;                                    // relu >= 0
#pragma unroll
        for (int v = 0; v < 8; ++v)
            mval = fmaxf(mval, fmaxf(acc0[v] * sc + sh, 0.f));
        if (hi == 0) {                                       // valid rows 16..19
#pragma unroll
            for (int v = 0; v < 4; ++v)
                mval = fmaxf(mval, fmaxf(acc1[v] * sc + sh, 0.f));
        }
        mval = fmaxf(mval, __shfl_xor(mval, 16, 32));        // combine halves
        if (lane < 16) out[(size_t)gpt * ldo + o] = mval;
    }
}

// ---------------------------------------------------------------------------
// 4) head GEMM with fused bias + BN + ReLU. Block = 4 waves; A row-tile
//    (16 x K) staged into LDS via async copies, shared by 4 col-tile waves.
// ---------------------------------------------------------------------------
template <int KDIM>
__global__ __launch_bounds__(128)
void gemm_bn_relu_kernel(const float* __restrict__ A, int lda,
                         const float* __restrict__ W,    // (Nout, KDIM)
                         const float* __restrict__ bias,
                         const float* __restrict__ g,
                         const float* __restrict__ be,
                         const float* __restrict__ m,
                         const float* __restrict__ vv,
                         float* __restrict__ out, int ldo) {
    constexpr int KS  = KDIM / 4;
    constexpr int LDA = KDIM + 4;                   // 16B-aligned rows
    __shared__ float As[16 * LDA];

    const int tid  = threadIdx.x;
    const int lane = tid & 31;
    const int wave = tid >> 5;
    const int r16  = lane & 15;
    const int hi   = lane >> 4;
    const int rowbase = blockIdx.x * 16;
    const int nbase   = blockIdx.y * 64 + wave * 16;

    for (int t = tid; t < 16 * (KDIM / 4); t += 128) {
        int rr = t / (KDIM / 4), k4 = (t % (KDIM / 4)) * 4;
        async_cp16(&As[rr * LDA + k4], A + (size_t)(rowbase + rr) * lda + k4);
    }
    async_cp_fence();
    __syncthreads();

    const int o = nbase + r16;
    const float* __restrict__ Wr = W + (size_t)o * KDIM;
    __builtin_prefetch(Wr, 0, 1);                   // global_prefetch_b8

    v8f acc = {};
#pragma unroll 8
    for (int s = 0; s < KS; ++s) {
        int k0 = s * 4 + hi * 2;
        v2f a  = *(const v2f*)&As[r16 * LDA + k0];  // 8B-aligned
        v2f bf = *(const v2f*)(Wr + k0);
        acc = wmma4(a, bf, acc);
    }
    const float sc = g[o] * rsqrtf(vv[o] + EPSV);
    const float sh = be[o] + (bias[o] - m[o]) * sc;
#pragma unroll
    for (int v = 0; v < 8; ++v) {
        float h = fmaxf(acc[v] * sc + sh, 0.f);
        out[(size_t)(rowbase + v + 8 * hi) * ldo + o] = h;
    }
}

// ---------------------------------------------------------------------------
// 5) final 256 -> 2 linear (too skinny for WMMA; trivial VALU dot)
// ---------------------------------------------------------------------------
__global__ void final_linear_kernel(const float* __restrict__ A,   // (rows,256)
                                    const float* __restrict__ W,   // (2,256)
                                    const float* __restrict__ bias,
                                    float* __restrict__ out, int rows) {
    int i   = blockIdx.x * blockDim.x + threadIdx.x;
    int row = i >> 1, cls = i & 1;
    if (row >= rows) return;
    const float* a = A + (size_t)row * 256;
    const float* w = W + (size_t)cls * 256;
    float s = bias[cls];
    for (int k = 0; k < 256; k += 4) {
        float4 av = *(const float4*)(a + k);
        float4 wv = *(const float4*)(w + k);
        s += av.x * wv.x + av.y * wv.y + av.z * wv.z + av.w * wv.w;
    }
    out[(size_t)row * 2 + cls] = s;
}

// ---------------------------------------------------------------------------
extern "C" void kernel_launch(void* const* d_in, const int* in_sizes, int n_in,
                              void* d_out, int out_size, void* d_ws, size_t ws_size,
                              hipStream_t stream) {
    (void)in_sizes; (void)n_in; (void)out_size; (void)ws_size;
    const int B = 4, N = 4096, BN = B * N;
    const float* x = (const float*)d_in[0];
    auto in = [&](int i) { return (const float*)d_in[i]; };

    // workspace layout (all regions fully overwritten each call)
    char*  ws  = (char*)d_ws;
    float* xc  = (float*)ws;                 size_t off = (size_t)BN * 512 * 4;  // (B*N,512)
    float* h1  = (float*)(ws + off);         off += (size_t)BN * 1024 * 4;       // (B*N,1024)
    float* h2  = (float*)(ws + off);         off += (size_t)BN * 256 * 4;        // (B*N,256)
    float* xx  = (float*)(ws + off);         off += (size_t)BN * 4;              // (B*N,)
    int*   idx = (int*)(ws + off);                                               // (B*N,20)

    dim3 knnGrid(N / 64, B);

    // ---- layer 1: C=3  -> O=64  (xc[:, 0:64]) ----
    rowsq_kernel<<<BN / 256, 256, 0, stream>>>(x, 3, 3, xx, BN);
    knn_kernel<3><<<knnGrid, 128, 0, stream>>>(x, 3, xx, idx, N);
    edgeconv_kernel<3, 64><<<BN, 32, 0, stream>>>(
        x, 3, idx, in(1), in(2), in(3), in(4), in(5), in(6), xc + 0, 512, N);

    // ---- layer 2: C=64 -> O=64  (xc[:, 64:128]) ----
    rowsq_kernel<<<BN / 256, 256, 0, stream>>>(xc + 0, 512, 64, xx, BN);
    knn_kernel<64><<<knnGrid, 128, 0, stream>>>(xc + 0, 512, xx, idx, N);
    edgeconv_kernel<64, 64><<<BN, 32, 0, stream>>>(
        xc + 0, 512, idx, in(7), in(8), in(9), in(10), in(11), in(12), xc + 64, 512, N);

    // ---- layer 3: C=64 -> O=128 (xc[:, 128:256]) ----
    rowsq_kernel<<<BN / 256, 256, 0, stream>>>(xc + 64, 512, 64, xx, BN);
    knn_kernel<64><<<knnGrid, 128, 0, stream>>>(xc + 64, 512, xx, idx, N);
    edgeconv_kernel<64, 128><<<BN, 32, 0, stream>>>(
        xc + 64, 512, idx, in(13), in(14), in(15), in(16), in(17), in(18), xc + 128, 512, N);

    // ---- layer 4: C=128 -> O=256 (xc[:, 256:512]) ----
    rowsq_kernel<<<BN / 256, 256, 0, stream>>>(xc + 128, 512, 128, xx, BN);
    knn_kernel<128><<<knnGrid, 128, 0, stream>>>(xc + 128, 512, xx, idx, N);
    edgeconv_kernel<128, 256><<<BN, 32, 0, stream>>>(
        xc + 128, 512, idx, in(19), in(20), in(21), in(22), in(23), in(24), xc + 256, 512, N);

    // ---- head: 512 -> 1024 -> 256 -> 2 ----
    gemm_bn_relu_kernel<512><<<dim3(BN / 16, 1024 / 64), 128, 0, stream>>>(
        xc, 512, in(25), in(26), in(27), in(28), in(29), in(30), h1, 1024);
    gemm_bn_relu_kernel<1024><<<dim3(BN / 16, 256 / 64), 128, 0, stream>>>(
        h1, 1024, in(31), in(32), in(33), in(34), in(35), in(36), h2, 256);
    final_linear_kernel<<<(BN * 2 + 255) / 256, 256, 0, stream>>>(
        h2, in(37), in(38), (float*)d_out, BN);
}